// DeformationTrackerBiFlowModel_55825984913618
// MI455X (gfx1250) — compile-verified
//
#include <hip/hip_runtime.h>

// MI455X / gfx1250, wave32. Memory-bound recurrence (≈131 MB @ 23.3 TB/s ≈ 5.6 us
// floor); per-step 5x12 and 12x2 GEMMs mapped onto V_WMMA_F32_16X16X4_F32 in a
// transposed formulation (batch = N/lane dimension) so the autoregressive
// feedback out(t) -> x(t+1) stays in-register with no cross-lane traffic.
// tanh uses the CDNA5 hardware transcendental (v_tanh_f32) when the builtin
// exists, else a branch-free v_exp_f32/v_rcp_f32 formulation — no libm, no
// divergent exec-mask churn inside the serial T loop.

typedef __attribute__((ext_vector_type(2))) float v2f;
typedef __attribute__((ext_vector_type(8))) float v8f;

#define B_TOTAL 65536
#define T_STEPS 100
#define HID     12

__device__ __forceinline__ float fast_tanh(float x) {
#if __has_builtin(__builtin_amdgcn_tanhf)
    return __builtin_amdgcn_tanhf(x);                 // v_tanh_f32 (TRANS32)
#elif __has_builtin(__builtin_amdgcn_tanh_f32)
    return __builtin_amdgcn_tanh_f32(x);              // alt builtin spelling
#else
    // Branch-free: tanh(x) = 1 - 2/(exp2(2*log2e*x) + 1).
    // x->+inf: exp2->inf, rcp->0 -> +1 ; x->-inf: exp2->0, rcp(1)=1 -> -1.
    float e = __builtin_amdgcn_exp2f(x * 2.8853900817779268f);
    float r = __builtin_amdgcn_rcpf(e + 1.0f);
    return __builtin_fmaf(-2.0f, r, 1.0f);
#endif
}

__global__ __launch_bounds__(256) void deform_rnn_wmma(
    const float* __restrict__ cp,   // (B, T, 2) control_point_input
    const float* __restrict__ fin,  // (B, T, 3) finger_input
    const float* __restrict__ Wr,   // (5, 12)   W_rnn
    const float* __restrict__ br,   // (12)      b_rnn
    const float* __restrict__ Wo,   // (14, 2)   W_out
    const float* __restrict__ bo,   // (2)       b_out
    float* __restrict__ out)        // (B, T, 2)
{
    const int  lane     = threadIdx.x & 31;
    const int  wave     = threadIdx.x >> 5;
    const int  tileBase = (blockIdx.x * 8 + wave) * 16;   // 16 batch cols / wave
    const bool lo       = lane < 16;
    const int  m        = lane & 15;        // A: row M ; B/D: batch column
    const int  col      = tileBase + m;     // batch element owned by this lane

    // ---------------- constant A operands (weights, transposed) --------------
    // A layout (16x4 f32): VGPR0 = K0(lanes0-15)|K2(lanes16-31), VGPR1 = K1|K3.
    // GEMM1: h^T(12x16) = Wr^T(12x5) * x^T(5x16), K split 4 + 1(padded)
    v2f a1a, a1b;
    a1a[0] = (m < HID) ? (lo ? Wr[0 * HID + m] : Wr[2 * HID + m]) : 0.f;
    a1a[1] = (m < HID) ? (lo ? Wr[1 * HID + m] : Wr[3 * HID + m]) : 0.f;
    a1b[0] = (m < HID && lo) ? Wr[4 * HID + m] : 0.f;
    a1b[1] = 0.f;
    // GEMM2: out^T(2x16) = Wo^T[:,2:14](2x12) * h^T(12x16), K split 4+4+4.
    // (Wo rows 0,1 = cp0 contribution, folded into the C accumulator below.)
    v2f a2a, a2b, a2c;
    a2a[0] = (m < 2) ? (lo ? Wo[2 * 2 + m] : Wo[4 * 2 + m]) : 0.f;
    a2a[1] = (m < 2) ? (lo ? Wo[3 * 2 + m] : Wo[5 * 2 + m]) : 0.f;
    a2b[0] = (m < 2) ? (lo ? Wo[6 * 2 + m] : Wo[8 * 2 + m]) : 0.f;
    a2b[1] = (m < 2) ? (lo ? Wo[7 * 2 + m] : Wo[9 * 2 + m]) : 0.f;
    a2c[0] = (m < 2) ? (lo ? Wo[10 * 2 + m] : Wo[12 * 2 + m]) : 0.f;
    a2c[1] = (m < 2) ? (lo ? Wo[11 * 2 + m] : Wo[13 * 2 + m]) : 0.f;

    // ---------------- constant C accumulators --------------------------------
    // C/D layout: VGPR v = row v (lanes0-15) | row v+8 (lanes16-31), N = lane%16.
    v8f c1;  // b_rnn broadcast into h rows 0..11
    #pragma unroll
    for (int v = 0; v < 8; ++v) {
        int row = lo ? v : v + 8;
        c1[v] = (row < HID) ? br[row] : 0.f;
    }
    const float cp0_0 = cp[(size_t)col * (T_STEPS * 2) + 0];
    const float cp0_1 = cp[(size_t)col * (T_STEPS * 2) + 1];
    v8f c2;  // b_out + cp0 @ W_out[0:2]  (constant over t)
    #pragma unroll
    for (int v = 0; v < 8; ++v) c2[v] = 0.f;
    c2[0] = lo ? (bo[0] + cp0_0 * Wo[0 * 2 + 0] + cp0_1 * Wo[1 * 2 + 0]) : 0.f;
    c2[1] = lo ? (bo[1] + cp0_0 * Wo[0 * 2 + 1] + cp0_1 * Wo[1 * 2 + 1]) : 0.f;

    // ---------------- recurrence over T --------------------------------------
    const float* finCol = fin + (size_t)col * (T_STEPS * 3);
    float*       outCol = out + (size_t)col * (T_STEPS * 2);
    float prev0 = cp0_0, prev1 = cp0_1;   // meaningful in lanes 0-15

    for (int t = 0; t < T_STEPS; ++t) {
        // speculative stream-ahead prefetch of this column's fin row
        __builtin_prefetch(&finCol[(t + 8) * 3], 0, 1);

        // per-lane fin loads (compiler emits one global_load_b96); each value
        // lands directly in the B-layout slot that needs it
        const float f0 = finCol[t * 3 + 0];   // hi half: K2 of B1a.v0
        const float f1 = finCol[t * 3 + 1];   // hi half: K3 of B1a.v1
        const float f2 = finCol[t * 3 + 2];   // lo half: K0 of B1b.v0

        // B layout (4x16 f32): VGPR0 = K0|K2, VGPR1 = K1|K3 (mirrors C/D striping)
        v2f b1a, b1b;
        b1a[0] = lo ? prev0 : f0;     // rows: prev_out0 | fin0
        b1a[1] = lo ? prev1 : f1;     // rows: prev_out1 | fin1
        b1b[0] = lo ? f2 : 0.f;       // rows: fin2 | pad
        b1b[1] = 0.f;

        v8f h = __builtin_amdgcn_wmma_f32_16x16x4_f32(
            false, a1a, false, b1a, (short)0, c1, false, false);
        h = __builtin_amdgcn_wmma_f32_16x16x4_f32(
            false, a1b, false, b1b, (short)0, h, false, false);

        // h rows 0..11 live in D layout; hw-tanh then regroup into B K-quads
        const float th0 = fast_tanh(h[0]), th1 = fast_tanh(h[1]);
        const float th2 = fast_tanh(h[2]), th3 = fast_tanh(h[3]);
        const float th4 = fast_tanh(h[4]), th5 = fast_tanh(h[5]);
        const float th6 = fast_tanh(h[6]), th7 = fast_tanh(h[7]);
        const float s0 = __shfl_xor(th0, 16, 32);  // rows 8,9 -> lo half
        const float s1 = __shfl_xor(th1, 16, 32);
        const float s2 = __shfl_xor(th2, 16, 32);  // rows 2,3 -> hi half
        const float s3 = __shfl_xor(th3, 16, 32);
        const float s6 = __shfl_xor(th6, 16, 32);  // rows 6,7 -> hi half
        const float s7 = __shfl_xor(th7, 16, 32);

        v2f b2a, b2b, b2c;
        b2a[0] = lo ? th0 : s2;   b2a[1] = lo ? th1 : s3;   // rows h0,h1 | h2,h3
        b2b[0] = lo ? th4 : s6;   b2b[1] = lo ? th5 : s7;   // rows h4,h5 | h6,h7
        b2c[0] = lo ? s0 : th2;   b2c[1] = lo ? s1 : th3;   // rows h8,h9 | h10,h11

        v8f o = __builtin_amdgcn_wmma_f32_16x16x4_f32(
            false, a2a, false, b2a, (short)0, c2, false, false);
        o = __builtin_amdgcn_wmma_f32_16x16x4_f32(
            false, a2b, false, b2b, (short)0, o, false, false);
        o = __builtin_amdgcn_wmma_f32_16x16x4_f32(
            false, a2c, false, b2c, (short)0, o, false, false);

        prev0 = o[0];             // D rows 0,1 in lanes 0-15 feed next step's x
        prev1 = o[1];
        if (lo) {
            float2 r = make_float2(o[0], o[1]);
            *(float2*)&outCol[t * 2] = r;   // 8B-aligned store
        }
    }
}

extern "C" void kernel_launch(void* const* d_in, const int* in_sizes, int n_in,
                              void* d_out, int out_size, void* d_ws, size_t ws_size,
                              hipStream_t stream) {
    const float* cp  = (const float*)d_in[0];  // control_point_input (B,T,2)
    const float* fin = (const float*)d_in[1];  // finger_input       (B,T,3)
    const float* Wr  = (const float*)d_in[2];  // W_rnn (5,12)
    // d_in[3] = U_rnn — mathematically inert (h0 == 0 each call), unused
    const float* br  = (const float*)d_in[4];  // b_rnn (12)
    const float* Wo  = (const float*)d_in[5];  // W_out (14,2)
    const float* bo  = (const float*)d_in[6];  // b_out (2)
    float* out = (float*)d_out;                // (B,T,2)

    dim3 block(256);                        // 8 waves x 16 batch cols = 128/block
    dim3 grid(B_TOTAL / 128);               // 512 blocks, 4096 waves total
    hipLaunchKernelGGL(deform_rnn_wmma, grid, block, 0, stream,
                       cp, fin, Wr, br, Wo, bo, out);
}